// GATAggregator_42640435315326
// MI455X (gfx1250) — compile-verified
//
#include <hip/hip_runtime.h>
#include <hip/hip_bf16.h>
#include <math.h>

#define N_NODES   100000
#define N_EDGES   1600000
#define IN_FEATS  128
#define HEADS     4
#define OUT_F     16
#define HF        64            // HEADS*OUT_F
#define NEG_SLOPE 0.2f

typedef __attribute__((ext_vector_type(2))) float v2f;
typedef __attribute__((ext_vector_type(8))) float v8f;

// Order-preserving monotone map float -> uint32 so segment-max can use atomicMax(u32).
__device__ __forceinline__ unsigned ordEnc(float f) {
  unsigned u = __float_as_uint(f);
  return (u & 0x80000000u) ? ~u : (u | 0x80000000u);
}
__device__ __forceinline__ float ordDec(unsigned x) {
  return (x & 0x80000000u) ? __uint_as_float(x ^ 0x80000000u)
                           : __uint_as_float(~x);
}

__device__ __forceinline__ float lrelu(float x) {
  return x > 0.0f ? x : NEG_SLOPE * x;
}

// ---------------------------------------------------------------------------
// K1: z[N,64] = feats[N,128] @ W[64,128]^T  via V_WMMA_F32_16X16X4_F32 (fp32).
// block = 128 threads = 4 waves; block covers 16 node rows; wave w covers the
// 16 output channels of head w. K-loop: 32 x wmma 16x16x4.
// A 16x4 layout (ISA 7.12.2): lane%16 = M, VGPR j = K = 2*(lane/16)+j.
// B 4x16 mirrors A:           lane%16 = N, VGPR j = K = 2*(lane/16)+j.
// C/D: VGPR i -> M = i + 8*(lane/16), lane%16 = N.
// ---------------------------------------------------------------------------
__global__ __launch_bounds__(128) void k_project(const float* __restrict__ feats,
                                                 const float* __restrict__ W,
                                                 float* __restrict__ z) {
  const int t      = threadIdx.x & 31;
  const int w      = threadIdx.x >> 5;   // wave id == head id
  const int lane16 = t & 15;
  const int half   = t >> 4;
  const int nodeBase = blockIdx.x * 16;  // N_NODES % 16 == 0

  const float* __restrict__ arow = feats + (size_t)(nodeBase + lane16) * IN_FEATS + half * 2;
  const float* __restrict__ brow = W     + (size_t)(w * 16  + lane16) * IN_FEATS + half * 2;

  v8f acc = {};
#pragma unroll
  for (int kk = 0; kk < IN_FEATS; kk += 4) {
    v2f a = { arow[kk], arow[kk + 1] };
    v2f b = { brow[kk], brow[kk + 1] };
    acc = __builtin_amdgcn_wmma_f32_16x16x4_f32(false, a, false, b,
                                                (short)0, acc, false, false);
  }

  float* __restrict__ zo = z + (size_t)(nodeBase + half * 8) * HF + w * 16 + lane16;
#pragma unroll
  for (int i = 0; i < 8; ++i) zo[(size_t)i * HF] = acc[i];
}

// ---------------------------------------------------------------------------
// K2: el[n,h] = dot(z[n,h,:], attn_l[h,:]) ; er likewise. One thread per (n,h).
// ---------------------------------------------------------------------------
__global__ void k_logits(const float* __restrict__ z,
                         const float* __restrict__ attn_l,
                         const float* __restrict__ attn_r,
                         float* __restrict__ el, float* __restrict__ er) {
  int i = blockIdx.x * blockDim.x + threadIdx.x;
  if (i >= N_NODES * HEADS) return;
  int n = i >> 2, h = i & 3;
  const float* zp = z + (size_t)n * HF + h * OUT_F;
  const float* al = attn_l + h * OUT_F;
  const float* ar = attn_r + h * OUT_F;
  float sl = 0.0f, sr = 0.0f;
#pragma unroll
  for (int f = 0; f < OUT_F; ++f) {
    float v = zp[f];
    sl = fmaf(v, al[f], sl);
    sr = fmaf(v, ar[f], sr);
  }
  el[i] = sl;
  er[i] = sr;
}

// ---------------------------------------------------------------------------
// K3: init out=0, denom=0, m=ord(-inf).
// ---------------------------------------------------------------------------
__global__ void k_init(float* __restrict__ out, float* __restrict__ denom,
                       unsigned* __restrict__ menc) {
  int i = blockIdx.x * blockDim.x + threadIdx.x;
  if (i < N_NODES * HF) out[i] = 0.0f;
  if (i < N_NODES * HEADS) {
    denom[i] = 0.0f;
    menc[i]  = ordEnc(-INFINITY);
  }
}

// ---------------------------------------------------------------------------
// K4: segment max of leaky-relu logits over dst. One thread per (edge, head).
// ---------------------------------------------------------------------------
__global__ void k_edge_max(const int* __restrict__ src, const int* __restrict__ dst,
                           const float* __restrict__ el, const float* __restrict__ er,
                           unsigned* __restrict__ menc) {
  int i = blockIdx.x * blockDim.x + threadIdx.x;
  if (i >= N_EDGES * HEADS) return;
  int e = i >> 2, h = i & 3;
  int s = src[e], d = dst[e];
  float v = lrelu(el[s * HEADS + h] + er[d * HEADS + h]);
  atomicMax(&menc[d * HEADS + h], ordEnc(v));
}

// ---------------------------------------------------------------------------
// K5: ex = exp(e - m[dst]); denom[dst,h] += ex; out[dst,h,f] += ex * z[src,h,f].
// 64 threads per edge: one per (h,f). z and out are L2-resident (25.6 MB each).
// ---------------------------------------------------------------------------
__global__ __launch_bounds__(256) void k_edge_agg(const int* __restrict__ src,
                                                  const int* __restrict__ dst,
                                                  const float* __restrict__ el,
                                                  const float* __restrict__ er,
                                                  const unsigned* __restrict__ menc,
                                                  const float* __restrict__ z,
                                                  float* __restrict__ denom,
                                                  float* __restrict__ out) {
  long long i = (long long)blockIdx.x * blockDim.x + threadIdx.x;
  if (i >= (long long)N_EDGES * HF) return;
  int e = (int)(i >> 6);
  int r = (int)(i & 63);
  int h = r >> 4;
  int f = r & 15;
  int s = src[e], d = dst[e];
  float lg = lrelu(el[s * HEADS + h] + er[d * HEADS + h]);
  float m  = ordDec(menc[d * HEADS + h]);  // finite: this edge contributed to it
  float ex = __expf(lg - m);
  if (f == 0) atomicAdd(&denom[d * HEADS + h], ex);
  atomicAdd(&out[(size_t)d * HF + h * OUT_F + f],
            ex * z[(size_t)s * HF + h * OUT_F + f]);
}

// ---------------------------------------------------------------------------
// K6: out = out / denom + bias (in place); isolated dst nodes -> bias.
// ---------------------------------------------------------------------------
__global__ void k_final(float* __restrict__ out, const float* __restrict__ denom,
                        const float* __restrict__ bias) {
  int i = blockIdx.x * blockDim.x + threadIdx.x;
  if (i >= N_NODES * HF) return;
  int n = i >> 6;
  int c = i & 63;
  int h = c >> 4;
  float den = denom[n * HEADS + h];
  float v   = out[i];
  out[i] = (den != 0.0f) ? (v / den + bias[c]) : bias[c];
}

extern "C" void kernel_launch(void* const* d_in, const int* in_sizes, int n_in,
                              void* d_out, int out_size, void* d_ws, size_t ws_size,
                              hipStream_t stream) {
  (void)in_sizes; (void)n_in; (void)out_size; (void)ws_size;
  const float* feats  = (const float*)d_in[0];   // [N,128]
  const float* W      = (const float*)d_in[1];   // [64,128]
  const float* attn_l = (const float*)d_in[2];   // [4,16]
  const float* attn_r = (const float*)d_in[3];   // [4,16]
  const float* bias   = (const float*)d_in[4];   // [64]
  const int*   src    = (const int*)d_in[5];     // [E]
  const int*   dst    = (const int*)d_in[6];     // [E]
  float*       out    = (float*)d_out;           // [N,4,16]

  // Workspace layout (floats): z | el | er | m_enc(u32) | denom  (~32 MB)
  float*    ws    = (float*)d_ws;
  float*    z     = ws;
  float*    el    = z  + (size_t)N_NODES * HF;
  float*    er    = el + (size_t)N_NODES * HEADS;
  unsigned* menc  = (unsigned*)(er + (size_t)N_NODES * HEADS);
  float*    denom = (float*)(menc + (size_t)N_NODES * HEADS);

  // K1: WMMA projection (16 nodes x 64 channels per block)
  k_project<<<N_NODES / 16, 128, 0, stream>>>(feats, W, z);

  // K2: attention logits
  {
    int n = N_NODES * HEADS;
    k_logits<<<(n + 255) / 256, 256, 0, stream>>>(z, attn_l, attn_r, el, er);
  }

  // K3: init accumulators
  {
    int n = N_NODES * HF;
    k_init<<<(n + 255) / 256, 256, 0, stream>>>(out, denom, menc);
  }

  // K4: segment max over incoming edges
  {
    int n = N_EDGES * HEADS;
    k_edge_max<<<(n + 255) / 256, 256, 0, stream>>>(src, dst, el, er, menc);
  }

  // K5: exp / scatter-accumulate
  {
    long long n = (long long)N_EDGES * HF;
    k_edge_agg<<<(unsigned)((n + 255) / 256), 256, 0, stream>>>(src, dst, el, er,
                                                               menc, z, denom, out);
  }

  // K6: normalize + bias
  {
    int n = N_NODES * HF;
    k_final<<<(n + 255) / 256, 256, 0, stream>>>(out, denom, bias);
  }
}